// GATLayer_16767552323722
// MI455X (gfx1250) — compile-verified
//
#include <hip/hip_runtime.h>
#include <hip/hip_bf16.h>

typedef float v2f __attribute__((ext_vector_type(2)));
typedef float v8f __attribute__((ext_vector_type(8)));

#define ALPHA 0.2f
#define LOG2E 1.44269504f
#define NB 2048          // N nodes
#define CF 256           // channels
#define HH 4             // heads
#define CC 64            // channels per head
#define ITILE 64         // rows per workgroup in aggregation
#define JTILE 64         // j-tile width
#define PSTRIDE 65       // padded LDS stride for P (bank-conflict free A reads)

__device__ __forceinline__ float lrelu(float x) { return fmaxf(x, ALPHA * x); }

// ---------------------------------------------------------------------------
// Kernel 0: transpose W (C_OUT x C_IN) -> Wt (C_IN x C_OUT) for coalesced
// WMMA B-operand loads in the GEMM.
// ---------------------------------------------------------------------------
__global__ void transpose_W(const float* __restrict__ W, float* __restrict__ Wt) {
    int o = blockIdx.x;          // output channel
    int k = threadIdx.x;         // input channel
    Wt[k * CF + o] = W[o * CF + k];
}

// ---------------------------------------------------------------------------
// Kernel 1: f = X @ W^T + b     (M = B*N = 16384, N = 256, K = 256)
// One block = 16-row M tile staged in LDS; 16 waves each own a 16-col N tile.
// fp32 WMMA 16x16x4, K loop of 64 instructions.
// ---------------------------------------------------------------------------
__global__ __launch_bounds__(512) void gemm_f(const float* __restrict__ X,
                                              const float* __restrict__ Wt,
                                              const float* __restrict__ bias,
                                              float* __restrict__ f_out) {
    __shared__ float Xs[16 * 257];
    const int tid = threadIdx.x;
    const int m0  = blockIdx.x * 16;

    // stage X tile (16 x 256), coalesced
    for (int idx = tid; idx < 16 * 256; idx += 512) {
        int r = idx >> 8, cidx = idx & 255;
        Xs[r * 257 + cidx] = X[(size_t)(m0 + r) * CF + cidx];
    }
    __syncthreads();

    const int wv   = tid >> 5;
    const int lane = tid & 31;
    const int n0   = wv * 16;
    const int off  = (lane < 16) ? 0 : 2;   // K sub-row held by this half-wave
    const int nl   = lane & 15;

    v8f acc = {};
#pragma unroll 4
    for (int kc = 0; kc < CF; kc += 4) {
        v2f a, bv;
        a.x = Xs[nl * 257 + kc + off];       // A[m=nl][k=kc+off]
        a.y = Xs[nl * 257 + kc + off + 1];   // A[m=nl][k=kc+off+1]
        const float* wp = Wt + (size_t)(kc + off) * CF + n0 + nl;
        bv.x = wp[0];                        // B[k=kc+off][n=nl]
        bv.y = wp[CF];                       // B[k=kc+off+1][n=nl]
        acc = __builtin_amdgcn_wmma_f32_16x16x4_f32(
            false, a, false, bv, (short)0, acc, false, false);
    }

#pragma unroll
    for (int r = 0; r < 8; ++r) {
        int m   = m0 + r + ((lane >= 16) ? 8 : 0);
        int col = n0 + nl;
        f_out[(size_t)m * CF + col] = acc[r] + bias[col];
    }
}

// ---------------------------------------------------------------------------
// Kernel 2: s_src[m,h] = f[m,h,:].a_src[h]   s_dst[m,h] = f[m,h,:].a_dst[h]
// ---------------------------------------------------------------------------
__global__ void compute_s(const float* __restrict__ f,
                          const float* __restrict__ a,
                          float* __restrict__ s_src,
                          float* __restrict__ s_dst) {
    int gid = blockIdx.x * blockDim.x + threadIdx.x;   // < B*N*H = 65536
    int m = gid >> 2, h = gid & 3;
    const float* fr = f + (size_t)m * CF + h * CC;
    const float* as = a + h * (2 * CC);
    float ss = 0.f, sd = 0.f;
#pragma unroll 8
    for (int cidx = 0; cidx < CC; ++cidx) {
        float v = fr[cidx];
        ss += v * as[cidx];
        sd += v * as[CC + cidx];
    }
    s_src[gid] = ss;
    s_dst[gid] = sd;
}

// ---------------------------------------------------------------------------
// Kernel 3: fused masked-softmax attention aggregation.
// Grid: (N/ITILE, B). Block: 512 threads = 16 waves.
// Wave w owns (head = w>>2, 16-col tile ct = w&3) and 4 row-blocks of 16.
// All scores are kept in the exp2 domain (pre-scaled by log2e).
// ---------------------------------------------------------------------------
__global__ __launch_bounds__(512) void gat_agg(const float* __restrict__ f,
                                               const int* __restrict__ adj,
                                               const float* __restrict__ s_src,
                                               const float* __restrict__ s_dst,
                                               float* __restrict__ out) {
    extern __shared__ float smem[];
    float* sdst_s  = smem;                 // NB*HH = 8192   (log2e-scaled)
    float* ssrc_s  = sdst_s + NB * HH;     // ITILE*HH = 256 (log2e-scaled)
    float* mval_s  = ssrc_s + ITILE * HH;  // 256
    float* uv_s    = mval_s + ITILE * HH;  // 512: (ssrc-m, 0.2*ssrc-m) pairs
    float* P       = uv_s + 2 * ITILE * HH;     // HH*ITILE*PSTRIDE = 16640
    float* dpart   = P + HH * ITILE * PSTRIDE;  // 512
    float* denom_s = dpart + 512;               // 256

    const int tid  = threadIdx.x;
    const int bb   = blockIdx.y;
    const int i0   = blockIdx.x * ITILE;
    const int wv   = tid >> 5;
    const int lane = tid & 31;

    // ---- stage log2e-scaled s_dst (whole batch) and s_src (our i-tile) ----
    for (int idx = tid; idx < NB * HH; idx += 512)
        sdst_s[idx] = s_dst[(size_t)bb * NB * HH + idx] * LOG2E;
    if (tid < ITILE * HH)
        ssrc_s[tid] = s_src[(size_t)bb * NB * HH + i0 * HH + tid] * LOG2E;
    __syncthreads();

    // ---- pass 1: per-row masked max of s_dst (LR monotonic, scale > 0) ----
    for (int r = 0; r < 4; ++r) {
        int il = wv * 4 + r;
        const int* arow = adj + ((size_t)bb * NB + (i0 + il)) * NB;
        float m0 = -3.0e38f, m1 = -3.0e38f, m2 = -3.0e38f, m3 = -3.0e38f;
        for (int j = lane; j < NB; j += 32) {
            if (arow[j]) {
                const float* sp = &sdst_s[j * HH];
                m0 = fmaxf(m0, sp[0]); m1 = fmaxf(m1, sp[1]);
                m2 = fmaxf(m2, sp[2]); m3 = fmaxf(m3, sp[3]);
            }
        }
#pragma unroll
        for (int s = 16; s >= 1; s >>= 1) {
            m0 = fmaxf(m0, __shfl_xor(m0, s, 32));
            m1 = fmaxf(m1, __shfl_xor(m1, s, 32));
            m2 = fmaxf(m2, __shfl_xor(m2, s, 32));
            m3 = fmaxf(m3, __shfl_xor(m3, s, 32));
        }
        if (lane == 0) {
            mval_s[il * HH + 0] = lrelu(ssrc_s[il * HH + 0] + m0);
            mval_s[il * HH + 1] = lrelu(ssrc_s[il * HH + 1] + m1);
            mval_s[il * HH + 2] = lrelu(ssrc_s[il * HH + 2] + m2);
            mval_s[il * HH + 3] = lrelu(ssrc_s[il * HH + 3] + m3);
        }
    }
    __syncthreads();

    // precombine: lrelu(x)-m == max((ssrc-m)+sdst, (0.2*ssrc-m)+0.2*sdst)
    if (tid < ITILE * HH) {
        float ss = ssrc_s[tid], mv = mval_s[tid];
        uv_s[tid * 2 + 0] = ss - mv;
        uv_s[tid * 2 + 1] = ALPHA * ss - mv;
    }

    // ---- pass 2: stream j-tiles; P -> WMMA accumulate + denominator ----
    const int jl = tid & 63;   // j within tile (coalesced adj loads)
    const int g  = tid >> 6;   // combo group, 0..7 -> i rows g*8..g*8+7
    const int hW = wv >> 2;    // this wave's head
    const int ct = wv & 3;     // this wave's 16-col tile within head
    const int off = (lane < 16) ? 0 : 2;
    const int nl  = lane & 15;
    const int dcombo = tid >> 1;   // denominator owner combo (i_local*4 + h)
    const int dhalf  = tid & 1;

    const int* adj_base = adj + ((size_t)bb * NB + (i0 + g * 8)) * NB + jl;

    v8f acc[4] = {};
    float dsum = 0.f;

    for (int j0 = 0; j0 < NB; j0 += JTILE) {
        __syncthreads();   // previous tile fully consumed before overwrite

        // hoist the 4 per-head s_dst values for this thread's j column
        float sd[HH], se[HH];
#pragma unroll
        for (int h = 0; h < HH; ++h) {
            sd[h] = sdst_s[(j0 + jl) * HH + h];
            se[h] = ALPHA * sd[h];
        }

        // weights: w = adj ? exp2(max(u + sd, v + se)) : 0
#pragma unroll
        for (int ii = 0; ii < 8; ++ii) {
            int il = g * 8 + ii;
            int adjv = adj_base[(size_t)ii * NB + j0];
#pragma unroll
            for (int h = 0; h < HH; ++h) {
                int combo = il * HH + h;
                float u = uv_s[combo * 2 + 0];
                float v = uv_s[combo * 2 + 1];
                float x = fmaxf(u + sd[h], v + se[h]);
                float w = adjv ? __builtin_amdgcn_exp2f(x) : 0.f;
                P[(h * ITILE + il) * PSTRIDE + jl] = w;
            }
        }
        __syncthreads();

        // WMMA: out_tile(16x16) += P_tile(16x4) @ F_tile(4x16), 16 K-chunks
#pragma unroll 4
        for (int kc = 0; kc < 16; ++kc) {
            int jc = kc * 4;
            int jrow = j0 + jc + off;
            const float* fb = f + ((size_t)(bb * NB + jrow)) * CF
                                + hW * CC + ct * 16 + nl;
            v2f bv;
            bv.x = fb[0];     // F[j=jrow  ][col]
            bv.y = fb[CF];    // F[j=jrow+1][col]
#pragma unroll
            for (int rb = 0; rb < 4; ++rb) {
                int ir = rb * 16 + nl;
                v2f av;
                av.x = P[(hW * ITILE + ir) * PSTRIDE + jc + off];
                av.y = P[(hW * ITILE + ir) * PSTRIDE + jc + off + 1];
                acc[rb] = __builtin_amdgcn_wmma_f32_16x16x4_f32(
                    false, av, false, bv, (short)0, acc[rb], false, false);
            }
        }

        // denominator partials: combo owner sums its half of the P row
        {
            int il = dcombo >> 2, h = dcombo & 3;
            const float* pr = &P[(h * ITILE + il) * PSTRIDE + dhalf * 32];
            float s = 0.f;
#pragma unroll 8
            for (int q = 0; q < 32; ++q) s += pr[q];
            dsum += s;
        }
    }

    __syncthreads();
    dpart[tid] = dsum;                       // index == dcombo*2 + dhalf
    __syncthreads();
    if (tid < ITILE * HH)
        denom_s[tid] = dpart[tid * 2] + dpart[tid * 2 + 1];
    __syncthreads();

    // ---- normalize + store (16x16 f32 C/D lane layout) ----
#pragma unroll
    for (int rb = 0; rb < 4; ++rb) {
#pragma unroll
        for (int r = 0; r < 8; ++r) {
            int il = rb * 16 + r + ((lane >= 16) ? 8 : 0);
            float dn = denom_s[il * HH + hW];
            size_t m = (size_t)bb * NB + i0 + il;
            out[m * CF + hW * CC + ct * 16 + nl] = acc[rb][r] / dn;
        }
    }
}

// ---------------------------------------------------------------------------
extern "C" void kernel_launch(void* const* d_in, const int* in_sizes, int n_in,
                              void* d_out, int out_size, void* d_ws, size_t ws_size,
                              hipStream_t stream) {
    const float* node = (const float*)d_in[0];   // (8,2048,256) f32
    const int*   adj  = (const int*)  d_in[1];   // (8,2048,2048) i32
    const float* W    = (const float*)d_in[2];   // (256,256) f32
    const float* bias = (const float*)d_in[3];   // (256,) f32
    const float* a    = (const float*)d_in[4];   // (4,128) f32
    float* out = (float*)d_out;                  // (8,2048,256) f32

    float* f_ws = (float*)d_ws;                  // 4,194,304 floats
    float* Wt   = f_ws + (size_t)8 * NB * CF;    // 65,536 floats
    float* ssrc = Wt + CF * CF;                  // 65,536 floats
    float* sdst = ssrc + (size_t)8 * NB * HH;    // 65,536 floats

    transpose_W<<<CF, CF, 0, stream>>>(W, Wt);
    gemm_f<<<(8 * NB) / 16, 512, 0, stream>>>(node, Wt, bias, f_ws);
    compute_s<<<(8 * NB * HH) / 256, 256, 0, stream>>>(f_ws, a, ssrc, sdst);

    size_t smem = (size_t)(NB * HH + ITILE * HH * 2 + 2 * ITILE * HH
                           + HH * ITILE * PSTRIDE + 512 + ITILE * HH)
                  * sizeof(float);   // ~104 KB
    dim3 grid(NB / ITILE, 8);
    gat_agg<<<grid, 512, smem, stream>>>(f_ws, adj, ssrc, sdst, out);
}